// GraphNeuralNet_26534307954730
// MI455X (gfx1250) — compile-verified
//
#include <hip/hip_runtime.h>

typedef __attribute__((ext_vector_type(2))) float v2f;
typedef __attribute__((ext_vector_type(8))) float v8f;

#define WAVES_PER_BLOCK 8

// ---------------------------------------------------------------------------
// GEMM: C[M,Nc] = act(A[M,K] @ B[K,Nc] + bias), all f32 row-major.
// One wave32 computes a 16x64 strip of C: 4 WMMA tiles sharing one A fragment
// per k-step (4x arithmetic intensity on A, 4 independent WMMA dep-chains).
// V_WMMA_F32_16X16X4_F32 fragment layouts per CDNA5 ISA 7.12.2:
//   A 16x4:  lanes 0-15 M=0..15 {K=k,k+1}, lanes 16-31 M=0..15 {K=k+2,k+3}
//   B 4x16:  lanes 0-15 N=0..15 rows {k,k+1}, lanes 16-31 rows {k+2,k+3}
//   C 16x16: VGPR r -> M=r (lanes 0-15) / M=8+r (lanes 16-31), N=lane&15
// Requires Nc % 64 == 0 (true here: 128, 512).
// ---------------------------------------------------------------------------
__global__ __launch_bounds__(256)
void gemm_bias_act_wmma(const float* __restrict__ A, const float* __restrict__ B,
                        const float* __restrict__ bias, float* __restrict__ C,
                        int M, int K, int Nc, int act)
{
    const int wave = threadIdx.x >> 5;
    const int lane = threadIdx.x & 31;
    const int half = lane >> 4;
    const int l    = lane & 15;
    const int groupsN = Nc >> 6;                 // 64-wide strips
    const int tilesM  = (M + 15) >> 4;
    const int grp = blockIdx.x * WAVES_PER_BLOCK + wave;
    if (grp >= tilesM * groupsN) return;         // wave-uniform exit
    const int tm = grp / groupsN;
    const int tg = grp - tm * groupsN;
    const int m0 = tm << 4;
    const int n0 = tg << 6;

    // Clamp instead of predicating: unconditional coalesced loads, and the
    // polluted rows are exactly the rows the store loop skips.
    const int arow  = m0 + l;
    const int arowc = arow < M ? arow : M - 1;

    const v2f*   __restrict__ Ap = (const v2f*)(A + (size_t)arowc * K) + half;
    const float* __restrict__ Bp = B + (size_t)(half << 1) * Nc + n0 + l;
    const size_t bstep = (size_t)Nc << 2;        // 4 rows of B per k-step

    v8f acc0 = 0.f, acc1 = 0.f, acc2 = 0.f, acc3 = 0.f;
    for (int k = 0; k < K; k += 4) {
        v2f a = *Ap;                             // one b64 load: {K=k..k+1|k+2..k+3}
        v2f b0, b1, b2, b3;
        b0.x = Bp[0];   b0.y = Bp[Nc];
        b1.x = Bp[16];  b1.y = Bp[Nc + 16];
        b2.x = Bp[32];  b2.y = Bp[Nc + 32];
        b3.x = Bp[48];  b3.y = Bp[Nc + 48];
        acc0 = __builtin_amdgcn_wmma_f32_16x16x4_f32(false, a, false, b0, (short)0, acc0, false, false);
        acc1 = __builtin_amdgcn_wmma_f32_16x16x4_f32(false, a, false, b1, (short)0, acc1, false, false);
        acc2 = __builtin_amdgcn_wmma_f32_16x16x4_f32(false, a, false, b2, (short)0, acc2, false, false);
        acc3 = __builtin_amdgcn_wmma_f32_16x16x4_f32(false, a, false, b3, (short)0, acc3, false, false);
        Ap += 2;
        Bp += bstep;
    }

    auto store_tile = [&](v8f acc, int col) {
        const float bv = bias ? bias[col] : 0.0f;
#pragma unroll
        for (int r = 0; r < 8; ++r) {
            const int row = m0 + (half << 3) + r;
            if (row < M) {
                float v = acc[r] + bv;
                if (act) v = v > 0.0f ? v : (__expf(v) - 1.0f);   // ELU
                C[(size_t)row * Nc + col] = v;
            }
        }
    };
    store_tile(acc0, n0 + l);
    store_tile(acc1, n0 + 16 + l);
    store_tile(acc2, n0 + 32 + l);
    store_tile(acc3, n0 + 48 + l);
}

// ---------------------------------------------------------------------------
// Edge pass A: logits[e,h] = att[h] . leaky_relu(xl[src] + xr[dst]) and
// atomic segment-max over dst via order-preserving uint encoding.
// One wave per edge; each lane owns one float4 of a 128-wide head.
// ---------------------------------------------------------------------------
__global__ __launch_bounds__(256)
void gat_edge_logits(const float* __restrict__ xl, const float* __restrict__ xr,
                     const float* __restrict__ att, const int* __restrict__ srcA,
                     const int* __restrict__ dstA, int E, int Nn, int heads,
                     float* __restrict__ logits, unsigned* __restrict__ mEnc)
{
    const int wave = threadIdx.x >> 5;
    const int lane = threadIdx.x & 31;
    const long e = (long)blockIdx.x * WAVES_PER_BLOCK + wave;
    if (e >= (long)E + Nn) return;
    int s, d;
    if (e < E) { s = srcA[e]; d = dstA[e]; } else { s = d = (int)(e - E); }
    const size_t F = (size_t)heads << 7;
    const float4* __restrict__ xs = (const float4*)(xl + (size_t)s * F);
    const float4* __restrict__ xd = (const float4*)(xr + (size_t)d * F);
    const float4* __restrict__ at = (const float4*)att;
    for (int h = 0; h < heads; ++h) {
        const int idx = (h << 5) + lane;
        float4 a = xs[idx], b = xd[idx], w = at[idx];
        float e0 = a.x + b.x; e0 = e0 > 0.f ? e0 : 0.2f * e0;
        float e1 = a.y + b.y; e1 = e1 > 0.f ? e1 : 0.2f * e1;
        float e2 = a.z + b.z; e2 = e2 > 0.f ? e2 : 0.2f * e2;
        float e3 = a.w + b.w; e3 = e3 > 0.f ? e3 : 0.2f * e3;
        float p = e0 * w.x + e1 * w.y + e2 * w.z + e3 * w.w;
#pragma unroll
        for (int off = 16; off > 0; off >>= 1) p += __shfl_xor(p, off, 32);
        if (lane == 0) {
            logits[e * heads + h] = p;
            unsigned enc = (p >= 0.f) ? (__float_as_uint(p) | 0x80000000u)
                                      : ~__float_as_uint(p);
            atomicMax(mEnc + (size_t)d * heads + h, enc);
        }
    }
}

// Decode order-preserving uint max back to float (in place).
__global__ void decode_segmax(unsigned* __restrict__ m, int n)
{
    int i = blockIdx.x * blockDim.x + threadIdx.x;
    if (i >= n) return;
    unsigned u = m[i];
    float f = (u & 0x80000000u) ? __uint_as_float(u & 0x7fffffffu)
                                : __uint_as_float(~u);
    ((float*)m)[i] = f;
}

// Edge pass B: p = exp(logit - max[dst]); denom[dst] += p (atomic).
__global__ void gat_edge_expsum(float* __restrict__ logits, const float* __restrict__ m,
                                const int* __restrict__ dstA, int E, int Nn, int heads,
                                float* __restrict__ denom)
{
    long i = (long)blockIdx.x * blockDim.x + threadIdx.x;
    long tot = ((long)E + Nn) * heads;
    if (i >= tot) return;
    long e = i / heads;
    int h = (int)(i - e * heads);
    int d = (e < E) ? dstA[e] : (int)(e - E);
    float p = __expf(logits[i] - m[(size_t)d * heads + h]);
    logits[i] = p;
    atomicAdd(denom + (size_t)d * heads + h, p);
}

// Edge pass C: acc[dst] += (p/denom[dst]) * xl[src]  (512 or 128 f32 atomics).
__global__ __launch_bounds__(256)
void gat_edge_aggregate(const float* __restrict__ xl, const float* __restrict__ pnum,
                        const float* __restrict__ denom, const int* __restrict__ srcA,
                        const int* __restrict__ dstA, int E, int Nn, int heads,
                        float* __restrict__ acc)
{
    const int wave = threadIdx.x >> 5;
    const int lane = threadIdx.x & 31;
    const long e = (long)blockIdx.x * WAVES_PER_BLOCK + wave;
    if (e >= (long)E + Nn) return;
    int s, d;
    if (e < E) { s = srcA[e]; d = dstA[e]; } else { s = d = (int)(e - E); }
    const size_t F = (size_t)heads << 7;
    const float4* __restrict__ xs = (const float4*)(xl + (size_t)s * F);
    float* __restrict__ out = acc + (size_t)d * F;
    for (int h = 0; h < heads; ++h) {
        float alpha = pnum[e * heads + h] / (denom[(size_t)d * heads + h] + 1e-16f);
        float4 v = xs[(h << 5) + lane];
        float* o = out + (h << 7) + (lane << 2);
        atomicAdd(o + 0, alpha * v.x);
        atomicAdd(o + 1, alpha * v.y);
        atomicAdd(o + 2, alpha * v.z);
        atomicAdd(o + 3, alpha * v.w);
    }
}

// h = elu(acc + bias) elementwise, in place.
__global__ void bias_elu_inplace(float* __restrict__ a, const float* __restrict__ bias,
                                 long total, int F)
{
    long i = (long)blockIdx.x * blockDim.x + threadIdx.x;
    if (i >= total) return;
    float v = a[i] + bias[(int)(i % F)];
    a[i] = v > 0.f ? v : (__expf(v) - 1.f);
}

// logits = h2 @ w_out + b_out (CLASSES=2), then log_softmax. One thread/node.
__global__ void out_logsoftmax(const float* __restrict__ h, const float* __restrict__ w,
                               const float* __restrict__ b, float* __restrict__ out, int Nn)
{
    int i = blockIdx.x * blockDim.x + threadIdx.x;
    if (i >= Nn) return;
    const float* row = h + (size_t)i * 128;
    float l0 = b[0], l1 = b[1];
#pragma unroll 4
    for (int k = 0; k < 128; ++k) {
        float v = row[k];
        l0 += v * w[2 * k];
        l1 += v * w[2 * k + 1];
    }
    float mx = fmaxf(l0, l1);
    float lse = mx + __logf(__expf(l0 - mx) + __expf(l1 - mx));
    out[2 * i]     = l0 - lse;
    out[2 * i + 1] = l1 - lse;
}

extern "C" void kernel_launch(void* const* d_in, const int* in_sizes, int n_in,
                              void* d_out, int out_size, void* d_ws, size_t ws_size,
                              hipStream_t stream)
{
    const float* x     = (const float*)d_in[0];
    const int*   eidx  = (const int*)d_in[1];
    const float* w_in  = (const float*)d_in[2];
    const float* b_in  = (const float*)d_in[3];
    const float* wl1   = (const float*)d_in[4];
    const float* wr1   = (const float*)d_in[5];
    const float* att1  = (const float*)d_in[6];
    const float* b1    = (const float*)d_in[7];
    const float* wl2   = (const float*)d_in[8];
    const float* wr2   = (const float*)d_in[9];
    const float* att2  = (const float*)d_in[10];
    const float* b2    = (const float*)d_in[11];
    const float* w_out = (const float*)d_in[12];
    const float* b_out = (const float*)d_in[13];
    float* out = (float*)d_out;
    (void)n_in; (void)out_size; (void)ws_size;

    const int N  = in_sizes[0] / 128;   // 50000 nodes
    const int E  = in_sizes[1] / 2;     // 800000 edges
    const int Et = E + N;               // + self loops
    const int* srcA = eidx;
    const int* dstA = eidx + E;

    char* ws = (char*)d_ws;
    size_t off = 0;
    auto alloc = [&](size_t bytes) -> char* {
        char* p = ws + off;
        off = (off + bytes + 255) & ~(size_t)255;
        return p;
    };
    float*    h0   = (float*)alloc((size_t)N * 128 * 4);
    float*    xl1  = (float*)alloc((size_t)N * 512 * 4);
    float*    xr1  = (float*)alloc((size_t)N * 512 * 4);
    float*    acc1 = (float*)alloc((size_t)N * 512 * 4);   // becomes h1
    float*    xl2  = (float*)alloc((size_t)N * 128 * 4);
    float*    xr2  = (float*)alloc((size_t)N * 128 * 4);
    float*    acc2 = (float*)alloc((size_t)N * 128 * 4);   // becomes h2
    float*    lg1  = (float*)alloc((size_t)Et * 4 * 4);
    unsigned* m1   = (unsigned*)alloc((size_t)N * 4 * 4);
    float*    d1   = (float*)alloc((size_t)N * 4 * 4);
    float*    lg2  = (float*)alloc((size_t)Et * 4);
    unsigned* m2   = (unsigned*)alloc((size_t)N * 4);
    float*    d2   = (float*)alloc((size_t)N * 4);

    hipMemsetAsync(m1,   0, (size_t)N * 4 * 4,   stream);
    hipMemsetAsync(d1,   0, (size_t)N * 4 * 4,   stream);
    hipMemsetAsync(acc1, 0, (size_t)N * 512 * 4, stream);
    hipMemsetAsync(m2,   0, (size_t)N * 4,       stream);
    hipMemsetAsync(d2,   0, (size_t)N * 4,       stream);
    hipMemsetAsync(acc2, 0, (size_t)N * 128 * 4, stream);

    auto gemmGrid = [&](int M, int Nc) {
        int waves = ((M + 15) / 16) * (Nc / 64);   // one wave per 16x64 strip
        return (waves + WAVES_PER_BLOCK - 1) / WAVES_PER_BLOCK;
    };
    const int edgeBlocks = (Et + WAVES_PER_BLOCK - 1) / WAVES_PER_BLOCK;

    // Input MLP: h0 = elu(x @ w_in + b_in)
    gemm_bias_act_wmma<<<gemmGrid(N, 128), 256, 0, stream>>>(x, w_in, b_in, h0, N, 128, 128, 1);

    // ---- GATv2 layer 1 (heads=4, concat) ----
    gemm_bias_act_wmma<<<gemmGrid(N, 512), 256, 0, stream>>>(h0, wl1, nullptr, xl1, N, 128, 512, 0);
    gemm_bias_act_wmma<<<gemmGrid(N, 512), 256, 0, stream>>>(h0, wr1, nullptr, xr1, N, 128, 512, 0);
    gat_edge_logits<<<edgeBlocks, 256, 0, stream>>>(xl1, xr1, att1, srcA, dstA, E, N, 4, lg1, m1);
    decode_segmax<<<(N * 4 + 255) / 256, 256, 0, stream>>>(m1, N * 4);
    {
        long tot = (long)Et * 4;
        gat_edge_expsum<<<(int)((tot + 255) / 256), 256, 0, stream>>>(lg1, (const float*)m1, dstA, E, N, 4, d1);
    }
    gat_edge_aggregate<<<edgeBlocks, 256, 0, stream>>>(xl1, lg1, d1, srcA, dstA, E, N, 4, acc1);
    bias_elu_inplace<<<(int)(((long)N * 512 + 255) / 256), 256, 0, stream>>>(acc1, b1, (long)N * 512, 512);

    // ---- GATv2 layer 2 (heads=1, mean==identity) ----
    gemm_bias_act_wmma<<<gemmGrid(N, 128), 256, 0, stream>>>(acc1, wl2, nullptr, xl2, N, 512, 128, 0);
    gemm_bias_act_wmma<<<gemmGrid(N, 128), 256, 0, stream>>>(acc1, wr2, nullptr, xr2, N, 512, 128, 0);
    gat_edge_logits<<<edgeBlocks, 256, 0, stream>>>(xl2, xr2, att2, srcA, dstA, E, N, 1, lg2, m2);
    decode_segmax<<<(N + 255) / 256, 256, 0, stream>>>(m2, N);
    gat_edge_expsum<<<(Et + 255) / 256, 256, 0, stream>>>(lg2, (const float*)m2, dstA, E, N, 1, d2);
    gat_edge_aggregate<<<edgeBlocks, 256, 0, stream>>>(xl2, lg2, d2, srcA, dstA, E, N, 1, acc2);
    bias_elu_inplace<<<(int)(((long)N * 128 + 255) / 256), 256, 0, stream>>>(acc2, b2, (long)N * 128, 128);

    // Output head + log_softmax
    out_logsoftmax<<<(N + 255) / 256, 256, 0, stream>>>(acc2, w_out, b_out, out, N);
}